// CNN2D_37873021616665
// MI455X (gfx1250) — compile-verified
//
#include <hip/hip_runtime.h>
#include <hip/hip_bf16.h>
#include <math.h>

typedef __attribute__((ext_vector_type(2))) float v2f;
typedef __attribute__((ext_vector_type(8))) float v8f;

#define B_   16
#define C_   64
#define H_   224
#define W_   224
#define NK_  4
#define POOL_ 7
#define HW49 49
#define KTOT 576   // 64 ic * 9 taps (K reordered tap-major: kk' = t9*64 + ic)

// ---------------- workspace layout (floats) ----------------
#define WS_CTX    0         // [B][64][49]
#define WS_ZOOM   50176     // [B][64][49]
#define WS_AW     100352    // [B][4]
#define WS_BOXES  100416    // [B][4]
#define WS_AGGW   100480    // [B][576][64]  K-major (tap-major K) for WMMA B

// ============ Kernel 1: adaptive 7x7 average pool (memory-bound, 205MB) ============
// One wave per pooled output (32x32 block). Each lane loads 8 float4 (b128),
// wave32 shfl_xor reduction. 8 waves / 256-thread block.
__global__ __launch_bounds__(256) void k_pool(const float* __restrict__ x,
                                              float* __restrict__ ctx) {
    const int lane = threadIdx.x & 31;
    const int wave = threadIdx.x >> 5;
    const int o = blockIdx.x * 8 + wave;          // output index in [0, B*C*49)
    const int bc = o / HW49;
    const int p  = o % HW49;
    const int oh = p / POOL_, ow = p % POOL_;
    const float* base = x + (size_t)bc * (H_ * W_) + (size_t)(oh * 32) * W_ + ow * 32;
    float acc = 0.f;
#pragma unroll
    for (int i = 0; i < 8; ++i) {
        int q = i * 32 + lane;        // float4 index within 32x32 block
        int row = q >> 3;             // 8 float4 per row
        int c4 = q & 7;
        const float4 v = *(const float4*)(base + (size_t)row * W_ + c4 * 4);
        acc += (v.x + v.y) + (v.z + v.w);
    }
#pragma unroll
    for (int m = 16; m >= 1; m >>= 1) acc += __shfl_xor(acc, m, 32);
    if (lane == 0) ctx[o] = acc * (1.0f / 1024.0f);
}

// ============ Kernel 2: attention + roi-bbox head (one block per batch) ============
__global__ __launch_bounds__(256) void k_head(const float* __restrict__ ctx_ws,
                                              const float* __restrict__ fc1_w,
                                              const float* __restrict__ fc2_w,
                                              const float* __restrict__ fc2_b,
                                              const float* __restrict__ ln_g,
                                              const float* __restrict__ ln_b,
                                              const float* __restrict__ roi_w1,
                                              const float* __restrict__ roi_w2,
                                              const float* __restrict__ roi_b2,
                                              float* __restrict__ aw_ws,
                                              float* __restrict__ boxes_ws) {
    const int b = blockIdx.x;
    const int tid = threadIdx.x;
    __shared__ float ctx[C_ * HW49];
    __shared__ float a1[16 * HW49];
    __shared__ float r1[16 * HW49];
    __shared__ float f[NK_ * HW49];
    __shared__ float s[NK_ * HW49];
    __shared__ float rmean[16];

    for (int i = tid; i < C_ * HW49; i += 256) ctx[i] = ctx_ws[b * C_ * HW49 + i];
    __syncthreads();

    // fc1: 1x1 conv 64->16 + relu
    for (int i = tid; i < 16 * HW49; i += 256) {
        int oc = i / HW49, p = i % HW49;
        float acc = 0.f;
        for (int ic = 0; ic < C_; ++ic) acc += ctx[ic * HW49 + p] * fc1_w[oc * C_ + ic];
        a1[i] = fmaxf(acc, 0.f);
    }
    __syncthreads();

    // fc2: 3x3 conv 16->4, pad 1, + bias -> flat f[nk*49+p]
    for (int i = tid; i < NK_ * HW49; i += 256) {
        int nk = i / HW49, p = i % HW49, ph = p / 7, pw = p % 7;
        float acc = fc2_b[nk];
        for (int ic = 0; ic < 16; ++ic)
            for (int kh = 0; kh < 3; ++kh) {
                int iy = ph + kh - 1;
                if (iy < 0 || iy >= 7) continue;
                for (int kw = 0; kw < 3; ++kw) {
                    int ix = pw + kw - 1;
                    if (ix < 0 || ix >= 7) continue;
                    acc += a1[ic * HW49 + iy * 7 + ix] * fc2_w[nk * 144 + ic * 9 + kh * 3 + kw];
                }
            }
        f[i] = acc;
    }
    __syncthreads();

    // LayerNorm over groups of 4 consecutive flat elements
    if (tid < HW49) {
        float v0 = f[4 * tid], v1 = f[4 * tid + 1], v2 = f[4 * tid + 2], v3 = f[4 * tid + 3];
        float mu = 0.25f * (v0 + v1 + v2 + v3);
        float d0 = v0 - mu, d1 = v1 - mu, d2 = v2 - mu, d3 = v3 - mu;
        float var = 0.25f * (d0 * d0 + d1 * d1 + d2 * d2 + d3 * d3);
        float rs = rsqrtf(var + 1e-5f);
        f[4 * tid + 0] = d0 * rs * ln_g[0] + ln_b[0];
        f[4 * tid + 1] = d1 * rs * ln_g[1] + ln_b[1];
        f[4 * tid + 2] = d2 * rs * ln_g[2] + ln_b[2];
        f[4 * tid + 3] = d3 * rs * ln_g[3] + ln_b[3];
    }
    __syncthreads();

    // softmax(att/5) over the 49-axis, per column k
    if (tid < NK_) {
        int k = tid;
        float mx = -1e30f;
        for (int i = 0; i < HW49; ++i) mx = fmaxf(mx, f[4 * i + k] * 0.2f);
        float sum = 0.f;
        for (int i = 0; i < HW49; ++i) {
            float e = __expf(f[4 * i + k] * 0.2f - mx);
            s[4 * i + k] = e;
            sum += e;
        }
        float inv = 1.0f / sum;
        for (int i = 0; i < HW49; ++i) s[4 * i + k] *= inv;
    }
    __syncthreads();

    if (tid < NK_) {
        float acc = 0.f;
        for (int j = 0; j < HW49; ++j) acc += s[tid * HW49 + j];
        aw_ws[b * NK_ + tid] = acc;
    }

    // roi branch: 3x3 conv 64->16 + relu
    for (int i = tid; i < 16 * HW49; i += 256) {
        int oc = i / HW49, p = i % HW49, ph = p / 7, pw = p % 7;
        float acc = 0.f;
        for (int ic = 0; ic < C_; ++ic)
            for (int kh = 0; kh < 3; ++kh) {
                int iy = ph + kh - 1;
                if (iy < 0 || iy >= 7) continue;
                for (int kw = 0; kw < 3; ++kw) {
                    int ix = pw + kw - 1;
                    if (ix < 0 || ix >= 7) continue;
                    acc += ctx[ic * HW49 + iy * 7 + ix] * roi_w1[oc * 576 + ic * 9 + kh * 3 + kw];
                }
            }
        r1[i] = fmaxf(acc, 0.f);
    }
    __syncthreads();
    if (tid < 16) {
        float acc = 0.f;
        for (int p = 0; p < HW49; ++p) acc += r1[tid * HW49 + p];
        rmean[tid] = acc * (1.0f / 49.0f);
    }
    __syncthreads();
    if (tid < 4) {
        float v = roi_b2[tid];
        for (int ic = 0; ic < 16; ++ic) v += roi_w2[tid * 16 + ic] * rmean[ic];
        boxes_ws[b * 4 + tid] = v * 224.0f;
    }
}

// ============ Kernel 3: roi_align ============
__global__ __launch_bounds__(256) void k_roialign(const float* __restrict__ x,
                                                  const float* __restrict__ boxes,
                                                  float* __restrict__ zoomed) {
    int idx = blockIdx.x * 256 + threadIdx.x;
    if (idx >= B_ * C_ * HW49) return;
    int b = idx / (C_ * HW49);
    int p = idx % HW49;
    int py = p / 7, px = p % 7;
    float x1 = boxes[b * 4 + 0], y1 = boxes[b * 4 + 1];
    float x2 = boxes[b * 4 + 2], y2 = boxes[b * 4 + 3];
    float bw = fmaxf(x2 - x1, 1.0f) * (1.0f / POOL_);
    float bh = fmaxf(y2 - y1, 1.0f) * (1.0f / POOL_);
    const float* img = x + (size_t)(idx / HW49) * (H_ * W_);
    float acc = 0.f;
#pragma unroll
    for (int iy = 0; iy < 2; ++iy)
#pragma unroll
        for (int ix = 0; ix < 2; ++ix) {
            float offy = (float)py + (iy + 0.5f) * 0.5f;
            float offx = (float)px + (ix + 0.5f) * 0.5f;
            float yv = y1 + offy * bh;
            float xv = x1 + offx * bw;
            bool valid = (yv >= -1.0f) & (yv <= (float)H_) & (xv >= -1.0f) & (xv <= (float)W_);
            float y = fminf(fmaxf(yv, 0.f), (float)(H_ - 1));
            float xc = fminf(fmaxf(xv, 0.f), (float)(W_ - 1));
            int y0 = min((int)floorf(y), H_ - 1);
            int x0 = min((int)floorf(xc), W_ - 1);
            int y1i = min(y0 + 1, H_ - 1);
            int x1i = min(x0 + 1, W_ - 1);
            float ly = y - (float)y0, lx = xc - (float)x0;
            float hy = 1.f - ly, hx = 1.f - lx;
            float v = hy * hx * img[y0 * W_ + x0] + hy * lx * img[y0 * W_ + x1i] +
                      ly * hx * img[y1i * W_ + x0] + ly * lx * img[y1i * W_ + x1i];
            acc += valid ? v : 0.f;
        }
    zoomed[idx] = acc * 0.25f;
}

// ============ Kernel 4: aggregate weights, tap-major K: aggw[b][t9*64+ic][oc] ============
__global__ __launch_bounds__(256) void k_aggw(const float* __restrict__ aw,
                                              const float* __restrict__ weight,
                                              float* __restrict__ aggw) {
    int idx = blockIdx.x * 256 + threadIdx.x;
    if (idx >= B_ * KTOT * C_) return;
    int b = idx / (KTOT * C_);
    int rem = idx % (KTOT * C_);
    int kkp = rem / C_;           // kk' = t9*64 + ic
    int oc = rem % C_;
    int t9 = kkp >> 6;
    int ic = kkp & 63;
    float v = 0.f;
#pragma unroll
    for (int n = 0; n < NK_; ++n)
        v += aw[b * NK_ + n] * weight[n * (C_ * KTOT) + oc * KTOT + ic * 9 + t9];
    aggw[idx] = v;
}

// ============ Kernel 5: per-batch 49x576x64 GEMM via V_WMMA_F32_16X16X4_F32 ============
// K reordered tap-major so each 4-wide k-step shares one conv tap: the halo
// test + spatial offset are hoisted out of the (unrolled) 16-step inner loop.
__global__ __launch_bounds__(512) void k_conv_wmma(const float* __restrict__ zoomed,
                                                   const float* __restrict__ aggw,
                                                   const float* __restrict__ aw,
                                                   const float* __restrict__ bias,
                                                   float* __restrict__ out) {
    const int b = blockIdx.x;
    const int tid = threadIdx.x;
    __shared__ float z[C_ * HW49];
    for (int i = tid; i < C_ * HW49; i += 512) z[i] = zoomed[b * C_ * HW49 + i];
    __syncthreads();

    const int wave = tid >> 5;
    const int lane = tid & 31;
    const int mt = wave >> 2;            // position tile 0..3
    const int nt = wave & 3;             // channel tile 0..3
    const int lm = lane & 15;
    const bool hi = lane >= 16;
    const int m = mt * 16 + lm;          // A row (output position)
    const bool mvalid = m < HW49;
    const int ph = mvalid ? m / 7 : 0;
    const int pw = mvalid ? m % 7 : 0;
    const int n = nt * 16 + lm;          // B column (output channel)
    const int kofs = hi ? 2 : 0;         // A layout: lanes 16-31 hold K+2,K+3

    const float* Bm = aggw + (size_t)b * KTOT * C_ + n;
    v8f c = {0.f, 0.f, 0.f, 0.f, 0.f, 0.f, 0.f, 0.f};

    for (int t9 = 0; t9 < 9; ++t9) {     // conv tap (kh,kw)
        const int kh = t9 / 3;
        const int kw = t9 - 3 * kh;
        const int iy = ph + kh - 1;
        const int ix = pw + kw - 1;
        const bool ok = mvalid & (iy >= 0) & (iy < 7) & (ix >= 0) & (ix < 7);
        const int zoff = ok ? (iy * 7 + ix) : 0;     // safe LDS offset
        const float* Bt = Bm + (size_t)(t9 * 64) * C_;
#pragma unroll
        for (int icq = 0; icq < 16; ++icq) {         // 16 k-steps of 4 over ic
            const int ic0 = icq * 4 + kofs;
            v2f a, bb;
            float av0 = z[ic0 * HW49 + zoff];
            float av1 = z[(ic0 + 1) * HW49 + zoff];
            a[0] = ok ? av0 : 0.f;
            a[1] = ok ? av1 : 0.f;
            bb[0] = Bt[(size_t)ic0 * C_];
            bb[1] = Bt[(size_t)(ic0 + 1) * C_];
            c = __builtin_amdgcn_wmma_f32_16x16x4_f32(
                false, a, false, bb, (short)0, c, false, false);
        }
    }

    // agg bias (depends only on output channel n)
    float bv = 0.f;
#pragma unroll
    for (int q = 0; q < NK_; ++q) bv += aw[b * NK_ + q] * bias[q * C_ + n];

    // D layout: VGPR j = row j (lanes 0-15) / row j+8 (lanes 16-31)
#pragma unroll
    for (int j = 0; j < 8; ++j) {
        int row = mt * 16 + j + (hi ? 8 : 0);
        if (row < HW49)
            out[(size_t)b * C_ * HW49 + n * HW49 + row] = c[j] + bv;
    }
}

extern "C" void kernel_launch(void* const* d_in, const int* in_sizes, int n_in,
                              void* d_out, int out_size, void* d_ws, size_t ws_size,
                              hipStream_t stream) {
    const float* x      = (const float*)d_in[0];
    const float* fc1_w  = (const float*)d_in[1];
    const float* fc2_w  = (const float*)d_in[2];
    const float* fc2_b  = (const float*)d_in[3];
    const float* ln_g   = (const float*)d_in[4];
    const float* ln_b   = (const float*)d_in[5];
    const float* roi_w1 = (const float*)d_in[6];
    const float* roi_w2 = (const float*)d_in[7];
    const float* roi_b2 = (const float*)d_in[8];
    const float* weight = (const float*)d_in[9];
    const float* bias   = (const float*)d_in[10];
    float* out = (float*)d_out;
    float* ws  = (float*)d_ws;

    float* ctx    = ws + WS_CTX;
    float* zoomed = ws + WS_ZOOM;
    float* aw     = ws + WS_AW;
    float* boxes  = ws + WS_BOXES;
    float* aggw   = ws + WS_AGGW;

    // 1) adaptive pool 224x224 -> 7x7 (dominant 205MB read, b128 loads)
    k_pool<<<(B_ * C_ * HW49) / 8, 256, 0, stream>>>(x, ctx);
    // 2) attention + roi head
    k_head<<<B_, 256, 0, stream>>>(ctx, fc1_w, fc2_w, fc2_b, ln_g, ln_b,
                                   roi_w1, roi_w2, roi_b2, aw, boxes);
    // 3) roi_align
    k_roialign<<<(B_ * C_ * HW49 + 255) / 256, 256, 0, stream>>>(x, boxes, zoomed);
    // 4) aggregated weights (tap-major K layout)
    k_aggw<<<(B_ * KTOT * C_ + 255) / 256, 256, 0, stream>>>(aw, weight, aggw);
    // 5) final conv via V_WMMA_F32_16X16X4_F32 (144 k-steps, inner 16 unrolled)
    k_conv_wmma<<<B_, 512, 0, stream>>>(zoomed, aggw, aw, bias, out);
}